// Sister_4_k_squared_conv_61031485276438
// MI455X (gfx1250) — compile-verified
//
#include <hip/hip_runtime.h>
#include <hip/hip_bf16.h>
#include <stdint.h>

// ---------------------------------------------------------------------------
// Problem constants (from the reference)
// ---------------------------------------------------------------------------
#define B_IMG     8
#define CIN       1024
#define H_DIM     64
#define W_DIM     64
#define HW        (H_DIM * W_DIM)          // 4096 spatial positions
#define C_OUT     196                      // 4*7*7
#define C_OUT_PAD 208                      // pad to 13 * 16 for WMMA M tiles
#define N_ROI     1024
#define KPOOL     7
#define SAT_W     65                       // (H+1) x (W+1) integral image

typedef __attribute__((ext_vector_type(2))) float v2f;
typedef __attribute__((ext_vector_type(8))) float v8f;

// ---------------------------------------------------------------------------
// Kernel A: batch-sum the feature map.  F[c*HW + hw] = sum_b fm[b][c][hw].
// Roofline-dominant stage: 134 MB read + 16.8 MB write, coalesced float4.
// ---------------------------------------------------------------------------
__global__ void batchsum_kernel(const float* __restrict__ fm,
                                float* __restrict__ F) {
    const size_t NE = (size_t)CIN * HW;                  // 4,194,304 elements
    size_t idx = ((size_t)blockIdx.x * blockDim.x + threadIdx.x) * 4;
    if (idx >= NE) return;
    float4 s = make_float4(0.f, 0.f, 0.f, 0.f);
#pragma unroll
    for (int b = 0; b < B_IMG; ++b) {
        const float4 v = *(const float4*)(fm + (size_t)b * NE + idx);
        s.x += v.x; s.y += v.y; s.z += v.z; s.w += v.w;
    }
    *(float4*)(F + idx) = s;
}

// ---------------------------------------------------------------------------
// Kernel B: S(196x4096) = W(196x1024) * F(1024x4096) + B_IMG*bias, fp32 WMMA.
// One wave per 16x16 output tile, K marched in steps of 4 via
// V_WMMA_F32_16X16X4_F32.  Tiles: 13 (M) x 256 (N) = 3328 waves.
//
// Out-of-range M rows (196..207) are CLAMPED to row 195 instead of masked:
// the inner loop is branch-free (EXEC stays all-1s, as WMMA requires), and
// the duplicate rows land in padded S rows never read downstream.
//
// Fragment layouts per CDNA5 ISA 7.12.2 (wave32):
//   A 16x4 f32 : lane l, m = l&15; VGPR0/1 hold K = 2*(l>>4) + {0,1}
//   B 4x16 f32 : lane l, n = l&15; VGPR0/1 hold K = 2*(l>>4) + {0,1}
//   C/D 16x16  : lane l, n = l&15; VGPR v holds m = v + 8*(l>>4)
// ---------------------------------------------------------------------------
__global__ void gemm_wmma_kernel(const float* __restrict__ Wmat,   // [196][1024]
                                 const float* __restrict__ bias,   // [196]
                                 const float* __restrict__ F,      // [1024][4096]
                                 float* __restrict__ S) {          // [208][4096]
    const int flat  = blockIdx.x * blockDim.x + threadIdx.x;
    const int wave  = flat >> 5;
    const int lane  = threadIdx.x & 31;
    const int nTilesN = HW / 16;                 // 256
    const int mt = wave / nTilesN;               // 0..12
    const int nt = wave % nTilesN;               // 0..255
    if (mt >= C_OUT_PAD / 16) return;            // wave-uniform

    const int o0 = mt * 16;
    const int n0 = nt * 16;
    const int lhalf = lane >> 4;                 // 0 or 1
    const int l16   = lane & 15;

    // Accumulator initialised with B_IMG * bias broadcast along N (clamped).
    v8f acc;
#pragma unroll
    for (int v = 0; v < 8; ++v) {
        const int o = min(o0 + v + 8 * lhalf, C_OUT - 1);
        acc[v] = 8.0f * bias[o];
    }

    const int oA = min(o0 + l16, C_OUT - 1);     // clamp: no EXEC divergence
    const float* Wrow = Wmat + (size_t)oA * CIN;
    const float* Fcol = F + n0 + l16;

    for (int k0 = 0; k0 < CIN; k0 += 4) {
        const int kk = k0 + 2 * lhalf;
        const v2f a = *(const v2f*)(Wrow + kk);  // 8B-aligned b64 load
        v2f b;                                   // two coalesced b32 row loads
        b.x = Fcol[(size_t)kk * HW];
        b.y = Fcol[(size_t)(kk + 1) * HW];
        // 8-arg form: (neg_a, A, neg_b, B, c_mod, C, reuse_a, reuse_b)
        acc = __builtin_amdgcn_wmma_f32_16x16x4_f32(
            false, a, false, b, (short)0, acc, false, false);
    }

#pragma unroll
    for (int v = 0; v < 8; ++v) {
        const int m = v + 8 * lhalf;
        S[(size_t)(o0 + m) * HW + n0 + l16] = acc[v];
    }
}

// ---------------------------------------------------------------------------
// Kernel C: per-channel 2D inclusive prefix sum with zero pad -> sat[196][65][65]
// One 64-thread block per channel: row scans into LDS, then column scans out.
// ---------------------------------------------------------------------------
__global__ void integral_kernel(const float* __restrict__ S,
                                float* __restrict__ sat) {
    const int ch = blockIdx.x;                   // 0..195
    const int t  = threadIdx.x;                  // 0..63
    __shared__ float tile[H_DIM][W_DIM + 1];     // +1 pad: no bank conflicts

    const float* Sp = S + (size_t)ch * HW;
    float run = 0.0f;
    for (int x = 0; x < W_DIM; ++x) {            // thread t scans row t
        run += Sp[t * W_DIM + x];
        tile[t][x] = run;
    }
    __syncthreads();

    float* satp = sat + (size_t)ch * SAT_W * SAT_W;
    satp[t + 1]           = 0.0f;                // top pad row
    satp[(t + 1) * SAT_W] = 0.0f;                // left pad column
    if (t == 0) satp[0] = 0.0f;

    float crun = 0.0f;
    for (int y = 0; y < H_DIM; ++y) {            // thread t scans column t
        crun += tile[y][t];
        satp[(y + 1) * SAT_W + (t + 1)] = crun;
    }
}

// ---------------------------------------------------------------------------
// Kernel D: ROI pooling + mean.  One block per ROI; thread t == channel index
// ch = (jy*7+jx)*4 + c4 computes its bin via 4 SAT lookups; threads 0..3
// reduce the 49 bins per output channel.  out[i*4 + c4].
// ---------------------------------------------------------------------------
__global__ void roipool_kernel(const int* __restrict__ rois,
                               const float* __restrict__ sat,
                               float* __restrict__ out) {
    const int i = blockIdx.x;                    // ROI id
    const int t = threadIdx.x;                   // 0..255 (196 active)
    __shared__ float pool[C_OUT];

    const int* r = rois + i * 4;
    const int ymin = r[0] / 30, xmin = r[1] / 30;
    const int ymax = r[2] / 30, xmax = r[3] / 30;
    const int ystep = (ymax - ymin) / KPOOL;     // trunc == floor (non-negative)
    const int xstep = (xmax - xmin) / KPOOL;
    const float cnt = (float)(B_IMG * ystep * xstep);

    if (t < C_OUT) {
        const int bin = t >> 2;                  // jy*7 + jx
        const int jy = bin / KPOOL, jx = bin % KPOOL;
        const int ys = ymin + jy * ystep, ye = ys + ystep;
        const int xs = xmin + jx * xstep, xe = xs + xstep;
        const float* sp = sat + (size_t)t * SAT_W * SAT_W;   // ch == t
        const float bs = sp[ye * SAT_W + xe] - sp[ys * SAT_W + xe]
                       - sp[ye * SAT_W + xs] + sp[ys * SAT_W + xs];
        pool[t] = (cnt > 0.0f) ? bs / fmaxf(cnt, 1.0f) : 0.0f;
    }
    __syncthreads();

    if (t < 4) {
        float s = 0.0f;
#pragma unroll
        for (int b = 0; b < KPOOL * KPOOL; ++b) s += pool[t + 4 * b];
        out[i * 4 + t] = s / (float)(KPOOL * KPOOL);
    }
}

// ---------------------------------------------------------------------------
// Launch
// ---------------------------------------------------------------------------
extern "C" void kernel_launch(void* const* d_in, const int* in_sizes, int n_in,
                              void* d_out, int out_size, void* d_ws, size_t ws_size,
                              hipStream_t stream) {
    const float* fm    = (const float*)d_in[0];  // (8,1024,64,64)
    const int*   rois  = (const int*)  d_in[1];  // (1024,4)
    const float* convw = (const float*)d_in[2];  // (196,1024)
    const float* convb = (const float*)d_in[3];  // (196,)
    float*       out   = (float*)d_out;          // (1024,4,1,1) flat

    // Workspace layout (all fp32):
    //   F   : CIN * HW        = 16.78 MB
    //   S   : C_OUT_PAD * HW  =  3.41 MB
    //   sat : C_OUT * 65 * 65 =  3.31 MB
    char* ws = (char*)d_ws;
    float* F   = (float*)(ws);
    float* S   = (float*)(ws + (size_t)CIN * HW * sizeof(float));
    float* sat = (float*)(ws + ((size_t)CIN * HW + (size_t)C_OUT_PAD * HW) * sizeof(float));

    // A: batch sum (float4 per thread): 4,194,304/4 threads
    {
        const int threads = 256;
        const int blocks  = (int)(((size_t)CIN * HW / 4 + threads - 1) / threads); // 4096
        batchsum_kernel<<<blocks, threads, 0, stream>>>(fm, F);
    }
    // B: WMMA GEMM: 13 x 256 tiles = 3328 waves = 416 blocks of 8 waves
    {
        const int threads = 256;
        const int totalWaves = (C_OUT_PAD / 16) * (HW / 16);
        const int blocks = (totalWaves * 32 + threads - 1) / threads;              // 416
        gemm_wmma_kernel<<<blocks, threads, 0, stream>>>(convw, convb, F, S);
    }
    // C: integral images, one 64-thread block per channel
    integral_kernel<<<C_OUT, 64, 0, stream>>>(S, sat);

    // D: ROI pooling + mean, one block per ROI
    roipool_kernel<<<N_ROI, 256, 0, stream>>>(rois, sat, out);
}